// DETRHead_25821343384033
// MI455X (gfx1250) — compile-verified
//
#include <hip/hip_runtime.h>

#define BS 6
#define NQ 4096
#define C_DIM 256
#define NH 8
#define NL 4
#define NP 8
#define NZ 4
#define DIM 32
#define NV 19560   // 92*160 + 46*80 + 23*40 + 12*20

typedef __attribute__((ext_vector_type(2))) float v2f;
typedef __attribute__((ext_vector_type(8))) float v8f;

// ---------------------------------------------------------------------------
// GEMM: out[m, col0+n] = sum_k A[m,k] * W[n,k] + bias[n]
// A: (M,K) row-major, W: (N,K) row-major (B = W^T), out row-major ld=ldo.
// One wave computes a 64(M) x 16(N) tile with V_WMMA_F32_16X16X4_F32.
// K is compile-time; K-loop is software-pipelined with double-buffered
// register fragments so the 4 WMMAs overlap the next step's 5 loads.
// All addresses are uniform-base + int32 offsets (GVS saddr form).
// ---------------------------------------------------------------------------
template <int K>
__global__ void gemm_bt_wmma(const float* __restrict__ A,
                             const float* __restrict__ W,
                             const float* __restrict__ bias,
                             float* __restrict__ out,
                             int M, int N, int ldo, int col0)
{
    const int lane  = threadIdx.x & 31;
    const int wave  = blockIdx.x * (blockDim.x >> 5) + (threadIdx.x >> 5);
    const int ntile = N >> 4;
    const int mt    = wave / ntile;
    const int nt    = wave - mt * ntile;
    if (mt * 64 >= M) return;

    const int row0 = mt * 64;
    const int n    = nt * 16 + (lane & 15);     // output column within W
    const int kgrp = (lane >> 4) << 1;          // lanes 0-15 -> k0..1, 16-31 -> k2..3

    int aoff[4];                                // int32 element offsets (all < 2^25)
#pragma unroll
    for (int t = 0; t < 4; ++t) {
        int r = row0 + t * 16 + (lane & 15);
        if (r >= M) r = M - 1;                  // safe clamp; store is guarded
        aoff[t] = r * K + kgrp;
    }
    const int woff = n * K + kgrp;

    v8f acc[4];
    acc[0] = (v8f){}; acc[1] = (v8f){}; acc[2] = (v8f){}; acc[3] = (v8f){};

    // ---- prologue: fragments for k = 0
    v2f bf = *(const v2f*)(W + woff);
    v2f af[4];
#pragma unroll
    for (int t = 0; t < 4; ++t) af[t] = *(const v2f*)(A + aoff[t]);

    // ---- pipelined main loop
#pragma unroll 4
    for (int k = 4; k < K; k += 4) {
        v2f bfn = *(const v2f*)(W + woff + k);
        v2f afn[4];
#pragma unroll
        for (int t = 0; t < 4; ++t) afn[t] = *(const v2f*)(A + aoff[t] + k);
#pragma unroll
        for (int t = 0; t < 4; ++t)
            acc[t] = __builtin_amdgcn_wmma_f32_16x16x4_f32(
                false, af[t], false, bf, (short)0, acc[t], false, false);
        bf = bfn;
#pragma unroll
        for (int t = 0; t < 4; ++t) af[t] = afn[t];
    }
    // ---- epilogue: last k-step
#pragma unroll
    for (int t = 0; t < 4; ++t)
        acc[t] = __builtin_amdgcn_wmma_f32_16x16x4_f32(
            false, af[t], false, bf, (short)0, acc[t], false, false);

    const float bv   = bias[n];
    const int   mrow = (lane >> 4) << 3;        // C/D: lanes>=16 hold rows m+8..15
#pragma unroll
    for (int t = 0; t < 4; ++t) {
#pragma unroll
        for (int j = 0; j < 8; ++j) {
            int row = row0 + t * 16 + mrow + j;
            if (row < M) out[row * ldo + col0 + n] = acc[t][j] + bv;
        }
    }
}

// ---------------------------------------------------------------------------
// Sampling: one wave per (b, q, h); lane = channel d (DIM == 32 == wave32).
// Lane j owns sample j (l = j>>3, p = j&7): softmax of the 32 attn logits is
// a wave32 shuffle reduction; per-sample scalars are broadcast with __shfl;
// each corner gather is one coalesced 128-byte line from the L2-resident
// projected value (120 MB < 192 MB L2). Corner predicates are wave-uniform.
// ---------------------------------------------------------------------------
__global__ void msda_sample(const float* __restrict__ val,      // (BS, NV, 256)
                            const float* __restrict__ offattn,  // (BS*NQ, 768): [0,512)=off, [512,768)=attn logits
                            const float* __restrict__ refpts,   // (BS, NQ, NZ, 2)
                            float* __restrict__ out)            // (BS, NQ, 256)
{
    const int lane = threadIdx.x & 31;
    const int task = blockIdx.x * (blockDim.x >> 5) + (threadIdx.x >> 5);
    if (task >= BS * NQ * NH) return;
    const int h  = task & (NH - 1);
    const int bq = task >> 3;                 // b*NQ + q
    const int b  = bq / NQ;

    const int l = lane >> 3;                  // level of this lane's sample
    const int p = lane & 7;
    const int z = p & (NZ - 1);               // Z-anchor index

    const int Wl_[NL] = {160, 80, 40, 20};
    const int Hl_[NL] = { 92, 46, 23, 12};
    const int st_[NL] = {0, 14720, 18400, 19320};
    const int Wli = Wl_[l], Hli = Hl_[l], sli = st_[l];

    const unsigned oab = (unsigned)bq * 768u;
    const float offx  = offattn[oab + h * 64 + lane * 2 + 0];
    const float offy  = offattn[oab + h * 64 + lane * 2 + 1];
    const float logit = offattn[oab + 512 + h * 32 + lane];
    const float rx = refpts[((unsigned)bq * NZ + z) * 2 + 0];
    const float ry = refpts[((unsigned)bq * NZ + z) * 2 + 1];

    // pixel coords (align_corners=False):  (ref + off/W)*W - 0.5 = ref*W + off - 0.5
    const float x = rx * (float)Wli + offx - 0.5f;
    const float y = ry * (float)Hli + offy - 0.5f;

    // softmax over the 32 (level, point) logits of this (q, h)
    float mx = logit;
#pragma unroll
    for (int s = 16; s > 0; s >>= 1) mx = fmaxf(mx, __shfl_xor(mx, s, 32));
    float e = __expf(logit - mx);
    float sm = e;
#pragma unroll
    for (int s = 16; s > 0; s >>= 1) sm += __shfl_xor(sm, s, 32);
    const float w = e / sm;

    const float x0f = floorf(x), y0f = floorf(y);
    const float lx = x - x0f, ly = y - y0f;
    const int x0 = (int)x0f, y0 = (int)y0f;

    // uniform-base + 32-bit offset addressing: val[b*NV*256 + idx*256 + chan]
    const unsigned browbase = (unsigned)b * (NV * C_DIM);
    const unsigned chan     = (unsigned)(h * DIM + lane);

    float acc = 0.f;
    for (int j = 0; j < 32; ++j) {
        const int   Wj  = __shfl(Wli, j, 32);
        const int   Hj  = __shfl(Hli, j, 32);
        const int   sj  = __shfl(sli, j, 32);
        const int   x0j = __shfl(x0,  j, 32);
        const int   y0j = __shfl(y0,  j, 32);
        const float lxj = __shfl(lx,  j, 32);
        const float lyj = __shfl(ly,  j, 32);
        const float wj  = __shfl(w,   j, 32);

        const float w00 = (1.f - lxj) * (1.f - lyj) * wj;
        const float w10 = lxj * (1.f - lyj) * wj;
        const float w01 = (1.f - lxj) * lyj * wj;
        const float w11 = lxj * lyj * wj;

        // 4 corners; conditions are wave-uniform -> cheap scalar branches
        if (x0j >= 0 && x0j < Wj && y0j >= 0 && y0j < Hj)
            acc += w00 * val[browbase + (unsigned)(sj + y0j * Wj + x0j) * C_DIM + chan];
        if (x0j + 1 >= 0 && x0j + 1 < Wj && y0j >= 0 && y0j < Hj)
            acc += w10 * val[browbase + (unsigned)(sj + y0j * Wj + x0j + 1) * C_DIM + chan];
        if (x0j >= 0 && x0j < Wj && y0j + 1 >= 0 && y0j + 1 < Hj)
            acc += w01 * val[browbase + (unsigned)(sj + (y0j + 1) * Wj + x0j) * C_DIM + chan];
        if (x0j + 1 >= 0 && x0j + 1 < Wj && y0j + 1 >= 0 && y0j + 1 < Hj)
            acc += w11 * val[browbase + (unsigned)(sj + (y0j + 1) * Wj + x0j + 1) * C_DIM + chan];
    }

    out[(unsigned)bq * C_DIM + chan] = acc;
}

// ---------------------------------------------------------------------------
extern "C" void kernel_launch(void* const* d_in, const int* in_sizes, int n_in,
                              void* d_out, int out_size, void* d_ws, size_t ws_size,
                              hipStream_t stream)
{
    const float* query  = (const float*)d_in[0];
    const float* value  = (const float*)d_in[1];
    const float* refpts = (const float*)d_in[2];
    const float* W_off  = (const float*)d_in[3];
    const float* b_off  = (const float*)d_in[4];
    const float* W_attn = (const float*)d_in[5];
    const float* b_attn = (const float*)d_in[6];
    const float* W_val  = (const float*)d_in[7];
    const float* b_val  = (const float*)d_in[8];
    float* out = (float*)d_out;

    float* ws_val = (float*)d_ws;                         // (BS, NV, 256): 120.2 MB
    float* ws_oa  = ws_val + (size_t)BS * NV * C_DIM;     // (BS*NQ, 768):  75.5 MB

    // 1) value projection: (BS*NV, 256) @ W_val^T + b_val
    {
        const int M = BS * NV, N = C_DIM;
        const int waves  = ((M + 63) / 64) * (N / 16);
        const int blocks = (waves + 7) / 8;
        gemm_bt_wmma<C_DIM><<<blocks, 256, 0, stream>>>(value, W_val, b_val, ws_val,
                                                        M, N, C_DIM, 0);
    }
    // 2) sampling offsets: (BS*NQ, 256) @ W_off^T + b_off -> cols [0, 512)
    {
        const int M = BS * NQ, N = NH * NL * NP * 2;      // 512
        const int waves  = ((M + 63) / 64) * (N / 16);
        const int blocks = (waves + 7) / 8;
        gemm_bt_wmma<C_DIM><<<blocks, 256, 0, stream>>>(query, W_off, b_off, ws_oa,
                                                        M, N, 768, 0);
    }
    // 3) attention logits: (BS*NQ, 256) @ W_attn^T + b_attn -> cols [512, 768)
    {
        const int M = BS * NQ, N = NH * NL * NP;          // 256
        const int waves  = ((M + 63) / 64) * (N / 16);
        const int blocks = (waves + 7) / 8;
        gemm_bt_wmma<C_DIM><<<blocks, 256, 0, stream>>>(query, W_attn, b_attn, ws_oa,
                                                        M, N, 768, 512);
    }
    // 4) softmax + bilinear gather-accumulate
    {
        const int tasks  = BS * NQ * NH;                  // 196608 waves
        const int blocks = (tasks + 7) / 8;               // 8 waves / block
        msda_sample<<<blocks, 256, 0, stream>>>(ws_val, ws_oa, refpts, out);
    }
}